// SpectralContext_32375463477503
// MI455X (gfx1250) — compile-verified
//
#include <hip/hip_runtime.h>
#include <hip/hip_bf16.h>

#define N_   1024
#define C_   151
#define D_   4096
#define CPAD 160          // 151 padded to 10 tiles of 16
#define NMS_T 0.5f

typedef float v2f __attribute__((ext_vector_type(2)));
typedef float v8f __attribute__((ext_vector_type(8)));

// ---------------------------------------------------------------------------
// Kernel 1: fp32 WMMA GEMM  out[1024,151] = A[1024,4096] @ W[4096,151] + bias
// One wave per 32x16 output macro-tile (2 row-tiles sharing the B fragment):
// per K-step (K=4): 2x global_load_b64 (A) + 2x global_load_b32 (B) feed
// 2x v_wmma_f32_16x16x4_f32  ->  2 VMEM ops per WMMA instead of 3.
// Tail column tile handled by clamping n (garbage lanes never stored).
// ---------------------------------------------------------------------------
__global__ __launch_bounds__(128) void gemm_wmma_f32(
    const float* __restrict__ A, const float* __restrict__ W,
    const float* __restrict__ bias, float* __restrict__ out)
{
    const int wave = threadIdx.x >> 5;
    const int lane = threadIdx.x & 31;
    const int tile = blockIdx.x * 4 + wave;          // 320 macro-tiles total
    const int rowGrp  = tile / (CPAD / 16);          // 32 groups of 32 rows
    const int colTile = tile % (CPAD / 16);          // 10 column tiles

    const int kh  = lane >> 4;       // 0: lanes 0-15 (K=0,1)  1: lanes 16-31 (K=2,3)
    const int l15 = lane & 15;
    const int m0 = rowGrp * 32 + l15;                 // A row, tile 0
    const int m1 = m0 + 16;                           // A row, tile 1
    const int n  = colTile * 16 + l15;                // B column held by this lane
    const int nc = (n < C_) ? n : 0;                  // clamp keeps loads in-bounds

    const float* __restrict__ Arow0 = A + (size_t)m0 * D_ + 2 * kh;
    const float* __restrict__ Arow1 = A + (size_t)m1 * D_ + 2 * kh;
    const float* __restrict__ Wcol  = W + nc;

    v8f acc0 = {};
    v8f acc1 = {};
    for (int k0 = 0; k0 < D_; k0 += 4) {
        const int kb = k0 + 2 * kh;
        v2f b;
        b.x = Wcol[(size_t)kb * C_];                  // W[kb  ][n]
        b.y = Wcol[(size_t)(kb + 1) * C_];            // W[kb+1][n]
        v2f a0, a1;
        a0.x = Arow0[k0];                             // A[m0][k0 + 2*kh + 0]
        a0.y = Arow0[k0 + 1];
        a1.x = Arow1[k0];
        a1.y = Arow1[k0 + 1];
        // (neg_a, A, neg_b, B, c_mod, C, reuse_a, reuse_b)
        acc0 = __builtin_amdgcn_wmma_f32_16x16x4_f32(
            false, a0, false, b, (short)0, acc0, false, false);
        acc1 = __builtin_amdgcn_wmma_f32_16x16x4_f32(
            false, a1, false, b, (short)0, acc1, false, false);
    }

    if (n < C_) {
        const float bv = bias[n];
#pragma unroll
        for (int v = 0; v < 8; ++v) {
            const int r = v + 8 * kh;                 // C/D layout: M = v + 8*half
            out[(size_t)(rowGrp * 32 + r) * C_ + n]      = acc0[v] + bv;
            out[(size_t)(rowGrp * 32 + 16 + r) * C_ + n] = acc1[v] + bv;
        }
    }
}

// ---------------------------------------------------------------------------
// Kernel 2: row softmax -> probs (background column zeroed)
// ---------------------------------------------------------------------------
__global__ __launch_bounds__(256) void softmax_rows(
    const float* __restrict__ dists, float* __restrict__ probs)
{
    const int row = blockIdx.x * blockDim.x + threadIdx.x;
    if (row >= N_) return;
    const float* d = dists + (size_t)row * C_;
    float mx = -3.4e38f;
    for (int c = 0; c < C_; ++c) mx = fmaxf(mx, d[c]);
    float sum = 0.0f;
    for (int c = 0; c < C_; ++c) sum += __expf(d[c] - mx);
    const float inv = 1.0f / sum;
    float* p = probs + (size_t)row * C_;
    p[0] = 0.0f;                                      // background zeroed
    for (int c = 1; c < C_; ++c) p[c] = __expf(d[c] - mx) * inv;
}

// ---------------------------------------------------------------------------
// Kernel 3: greedy commit with on-the-fly IoU suppression.
// Single 1024-thread workgroup (32 wave32 waves). Per-row running argmax in
// LDS; global argmax = 10-step LDS tree reduction with flat-index tie-break.
// Avoids materializing the [N,N,C] IoU tensor (158M IoUs -> ~1M on the fly).
// ---------------------------------------------------------------------------
__global__ __launch_bounds__(1024) void greedy_nms(
    float* __restrict__ probs, const float* __restrict__ boxes,
    float* __restrict__ preds)
{
    __shared__ float sBest[N_];
    __shared__ int   sArg[N_];
    __shared__ float sRedV[N_];
    __shared__ int   sRedI[N_];

    const int j = threadIdx.x;

    // init per-row argmax (strict > keeps first-max semantics)
    {
        const float* p = probs + (size_t)j * C_;
        float bv = -3.4e38f; int ba = 0;
        for (int c = 0; c < C_; ++c) {
            const float v = p[c];
            if (v > bv) { bv = v; ba = c; }
        }
        sBest[j] = bv; sArg[j] = ba;
    }
    __syncthreads();

    for (int it = 0; it < N_; ++it) {
        // ---- global argmax (prefer lower flat index on ties, like jnp.argmax)
        sRedV[j] = sBest[j];
        sRedI[j] = j * C_ + sArg[j];
        __syncthreads();
        for (int s = 512; s > 0; s >>= 1) {
            if (j < s) {
                const float v1 = sRedV[j], v2 = sRedV[j + s];
                const int   i1 = sRedI[j], i2 = sRedI[j + s];
                if (v2 > v1 || (v2 == v1 && i2 < i1)) { sRedV[j] = v2; sRedI[j] = i2; }
            }
            __syncthreads();
        }
        const int flat = sRedI[0];
        const int b  = flat / C_;
        const int cl = flat - b * C_;
        __syncthreads();

        if (j == b) {
            preds[b] = (float)cl;        // commit
            sBest[b] = -1.0f;            // committed marker (probs are >= 0)
            sArg[b]  = 0;
        }

        // ---- suppression: IoU(box[j,cl], box[b,cl]) computed on the fly
        if (sBest[j] >= 0.0f) {
            const float* bj = boxes + ((size_t)j * C_ + cl) * 4;
            const float* bb = boxes + ((size_t)b * C_ + cl) * 4;
            const float x1 = fmaxf(bj[0], bb[0]);
            const float y1 = fmaxf(bj[1], bb[1]);
            const float x2 = fminf(bj[2], bb[2]);
            const float y2 = fminf(bj[3], bb[3]);
            const float iw = fmaxf(x2 - x1 + 1.0f, 0.0f);
            const float ih = fmaxf(y2 - y1 + 1.0f, 0.0f);
            const float inter = iw * ih;
            const float areaJ = (bj[2] - bj[0] + 1.0f) * (bj[3] - bj[1] + 1.0f);
            const float areaB = (bb[2] - bb[0] + 1.0f) * (bb[3] - bb[1] + 1.0f);
            const float uni = areaJ + areaB - inter;
            if (inter >= NMS_T * uni) {
                probs[(size_t)j * C_ + cl] = 0.0f;
                if (sArg[j] == cl) {     // dirtied our best -> rescan row (L2 hit)
                    const float* p = probs + (size_t)j * C_;
                    float bv = -3.4e38f; int ba = 0;
                    for (int c = 0; c < C_; ++c) {
                        const float v = (c == cl) ? 0.0f : p[c];
                        if (v > bv) { bv = v; ba = c; }
                    }
                    sBest[j] = bv; sArg[j] = ba;
                }
            }
        }
        __syncthreads();
    }
}

// ---------------------------------------------------------------------------
extern "C" void kernel_launch(void* const* d_in, const int* in_sizes, int n_in,
                              void* d_out, int out_size, void* d_ws, size_t ws_size,
                              hipStream_t stream)
{
    const float* obj_fmap = (const float*)d_in[0];   // [1024, 4096]
    const float* W_out    = (const float*)d_in[1];   // [4096, 151]
    const float* b_out    = (const float*)d_in[2];   // [151]
    const float* boxes    = (const float*)d_in[3];   // [1024, 151, 4]

    float* out   = (float*)d_out;                    // [1024*151] dists + [1024] preds
    float* probs = (float*)d_ws;                     // [1024*151] scratch

    const int macroTiles = (N_ / 32) * (CPAD / 16);  // 320 waves, 4 per block
    gemm_wmma_f32<<<macroTiles / 4, 128, 0, stream>>>(obj_fmap, W_out, b_out, out);
    softmax_rows<<<(N_ + 255) / 256, 256, 0, stream>>>(out, probs);
    greedy_nms<<<1, N_, 0, stream>>>(probs, boxes, out + (size_t)N_ * C_);
}